// GraphRestrictedBoltzmannMachine_3779571220570
// MI455X (gfx1250) — compile-verified
//
#include <hip/hip_runtime.h>
#include <hip/hip_bf16.h>
#include <stdint.h>

// Problem constants (match reference)
#define V   131072
#define H   8192
#define DEG 64
#define E   (H * DEG)
#define NN  (V + H)
#define B   128

typedef _Float16 half4 __attribute__((ext_vector_type(4)));
typedef _Float16 half8 __attribute__((ext_vector_type(8)));
typedef int      v4i   __attribute__((ext_vector_type(4)));
typedef __attribute__((address_space(3))) v4i* lds_v4i_p;

// ---------- CDNA5 async global->LDS helpers (guarded; safe fallbacks) ------
__device__ __forceinline__ void async_copy_b128(const void* gsrc, void* ldst) {
#if __has_builtin(__builtin_amdgcn_global_load_async_to_lds_b128)
  __builtin_amdgcn_global_load_async_to_lds_b128(
      (v4i*)gsrc,            // generic ptr to 16B vector (per diagnostic)
      (lds_v4i_p)ldst,       // LDS destination
      /*imm offset*/ 0, /*cpol*/ 0);
#else
  *reinterpret_cast<uint4*>(ldst) = *reinterpret_cast<const uint4*>(gsrc);
#endif
}

__device__ __forceinline__ void async_wait0() {
#if __has_builtin(__builtin_amdgcn_s_wait_asynccnt)
  __builtin_amdgcn_s_wait_asynccnt(0);
#else
  asm volatile("s_wait_asynccnt 0" ::: "memory");
#endif
}

// ---------------------------------------------------------------------------
// Kernel 1: pack + transpose.  s (B x NN f32, row-major) -> sT (V x B f16).
// s values are +-1 so f16 conversion is exact.  Tile = 64 v x 64 b.
// Phase 1 pulls the f32 tile straight into LDS with async b128 copies
// (no VGPR round trip); phase 2 converts and writes coalesced 16B chunks.
// ---------------------------------------------------------------------------
#define T1_STRIDE 68  // 64 f32 + 4 pad (272 B rows: 16B-aligned b128 LDS)

__global__ void k_pack_transpose(const float* __restrict__ s,
                                 _Float16* __restrict__ sT) {
  __shared__ __align__(16) float tile[64 * T1_STRIDE];

  const int v0 = blockIdx.x * 64;
  const int b0 = blockIdx.y * 64;
  const int t  = threadIdx.x;  // 256 threads

  // Phase 1: 64 rows (b) x 64 f32 = 1024 x 16B chunks, 4 per thread.
#pragma unroll
  for (int i = 0; i < 4; ++i) {
    int e  = i * 256 + t;        // 0..1023
    int bl = e >> 4;             // 0..63  (batch row in tile)
    int c  = e & 15;             // 0..15  (16B chunk in row)
    const float* g = s + (size_t)(b0 + bl) * NN + v0 + c * 4;
    async_copy_b128(g, &tile[bl * T1_STRIDE + c * 4]);
  }
  async_wait0();
  __syncthreads();

  // Phase 2: write sT rows (v-major).  Each thread emits one 16B chunk
  // (8 f16 = 8 consecutive batch values) per pass; 2 passes cover the tile.
#pragma unroll
  for (int p = 0; p < 2; ++p) {
    int e  = p * 256 + t;        // 0..511
    int vl = e >> 3;             // 0..63
    int bg = e & 7;              // 0..7  -> batch group of 8
    half8 h;
#pragma unroll
    for (int k = 0; k < 8; ++k)
      h[k] = (_Float16)tile[(bg * 8 + k) * T1_STRIDE + vl];
    *reinterpret_cast<half8*>(sT + (size_t)(v0 + vl) * B + b0 + bg * 8) = h;
  }
}

// ---------------------------------------------------------------------------
// Kernel 2: gather + segment reduce.
// Block = 256 threads (8 wave32), owns 32 consecutive hidden units.
// Each wave processes 4 hidden units; per edge d:
//   - adj/q read with uniform (SGPR) indices -> scalar loads, read ONCE total
//   - one global_load_b64 per wave fetches the full 256B sT row
//     (lane holds 4 batch values as f16), FMA into f32 accumulators.
// Results staged in LDS so the global store is coalesced 128B rows.
// ---------------------------------------------------------------------------
#define T2_STRIDE 132  // 128 f32 + 4 pad (528 B rows: 16B-aligned b128)

__global__ void k_field(const _Float16* __restrict__ sT,
                        const float* __restrict__ linear,
                        const float* __restrict__ quadratic,
                        const int* __restrict__ adj,
                        float* __restrict__ out) {
  __shared__ __align__(16) float tile[32 * T2_STRIDE];

  const int h0   = blockIdx.x * 32;
  const int lane = threadIdx.x & 31;
  const int wv   = __builtin_amdgcn_readfirstlane(threadIdx.x >> 5);  // uniform

#pragma unroll
  for (int i = 0; i < 4; ++i) {
    const int h  = h0 + wv * 4 + i;   // uniform per wave
    const int hl = h - h0;            // 0..31, unique per (wave,i)

    float a0 = 0.f, a1 = 0.f, a2 = 0.f, a3 = 0.f;
#pragma unroll 8
    for (int d = 0; d < DEG; ++d) {
      const int   v = adj[(size_t)h * DEG + d];        // SGPR load (uniform)
      const float w = quadratic[(size_t)h * DEG + d];  // SGPR load (uniform)
      // Whole 256B row fetched by one b64/wave: lane -> batch 4*lane..4*lane+3
      half4 g = reinterpret_cast<const half4*>(sT + (size_t)v * B)[lane];
      a0 = __builtin_fmaf(w, (float)g.x, a0);
      a1 = __builtin_fmaf(w, (float)g.y, a1);
      a2 = __builtin_fmaf(w, (float)g.z, a2);
      a3 = __builtin_fmaf(w, (float)g.w, a3);
    }
    const float lin = linear[V + h];                    // SGPR load (uniform)
    float4 r = make_float4(a0 + lin, a1 + lin, a2 + lin, a3 + lin);
    *reinterpret_cast<float4*>(&tile[hl * T2_STRIDE + lane * 4]) = r;
  }
  __syncthreads();

  // Flush: out[b][h0 + hl], lanes span hl (contiguous 128B stores per b-row).
#pragma unroll
  for (int i = 0; i < 16; ++i) {
    int idx = i * 256 + threadIdx.x;  // 0..4095 = 128 b x 32 hl
    int b   = idx >> 5;
    int hl  = idx & 31;
    out[(size_t)b * H + h0 + hl] = tile[hl * T2_STRIDE + b];
  }
}

// ---------------------------------------------------------------------------
extern "C" void kernel_launch(void* const* d_in, const int* in_sizes, int n_in,
                              void* d_out, int out_size, void* d_ws, size_t ws_size,
                              hipStream_t stream) {
  (void)in_sizes; (void)n_in; (void)out_size; (void)ws_size;

  const float* s         = (const float*)d_in[0];  // (B, NN) f32
  const float* linear    = (const float*)d_in[1];  // (NN,)   f32
  const float* quadratic = (const float*)d_in[2];  // (E,)    f32
  const int*   flat_adj  = (const int*)d_in[3];    // (E,)    i32
  // d_in[4..6] (flat_j_idx, segment_ids, hidden_idx) are identities; unused.

  float*    out = (float*)d_out;       // (B, H) f32
  _Float16* sT  = (_Float16*)d_ws;     // (V, B) f16 = 32 MB scratch

  // Kernel 1: transpose/pack s -> sT   (tiles: 2048 x 2)
  dim3 g1(V / 64, B / 64);
  k_pack_transpose<<<g1, 256, 0, stream>>>(s, sT);

  // Kernel 2: gather + reduce          (256 blocks x 32 hidden units)
  k_field<<<H / 32, 256, 0, stream>>>(sT, linear, quadratic, flat_adj, out);
}